// GraphSequenceModel_47931835023398
// MI455X (gfx1250) — compile-verified
//
#include <hip/hip_runtime.h>
#include <hip/hip_bf16.h>

// ---------------------------------------------------------------------------
// GraphSequenceModel on MI455X (gfx1250, wave32)
//
// Pipeline:
//   K1  xl = x @ W_gcn^T      (V_WMMA_F32_16X16X4_F32, 16 nodes/wave) + deg=1
//   K2  deg[dst] += 1 per edge (int4-vectorized edge loads, f32 atomics)
//   K3  dinv = rsqrt(deg)      (deg >= 1 always: self loops)
//   K4  gout = b_gcn + dinv^2 * xl               (self-loop term)
//   K5  gout[dst] += dinv[s]*dinv[d]*xl[src]     (edge scatter, f32 atomics)
//   K6  u = gout @ W_ih^T + (b_ih+b_hh)          (V_WMMA_F32_16X16X4_F32)
//   K7  sequential tanh recurrence over 160k steps (register-resident W_hh,
//       global_prefetch_b8 ahead of the u stream), relu->linear->sigmoid.
// ---------------------------------------------------------------------------

typedef __attribute__((ext_vector_type(2))) float v2f;
typedef __attribute__((ext_vector_type(8))) float v8f;

static constexpr int T_  = 8;
static constexpr int N_  = 20000;
static constexpr int E_  = 1280000;
static constexpr int TN_ = T_ * N_;        // 160000  (divisible by 128)
static constexpr long long TE_ = (long long)T_ * E_;  // 10,240,000

// 16x16x4 f32 WMMA wrapper: D = A(16x4) x B(4x16) + 0
__device__ inline v8f wmma_16x16x4(float a0, float a1, float b0, float b1) {
    v2f a; a.x = a0; a.y = a1;     // lane<16: A[M=lane][K=0..1]; lane>=16: K=2..3 (zero pad)
    v2f b; b.x = b0; b.y = b1;     // lane<16: B[K=0..1][N=lane]; lane>=16: K=2..3 (zero pad)
    v8f c = {};
    c = __builtin_amdgcn_wmma_f32_16x16x4_f32(
        /*neg_a=*/false, a, /*neg_b=*/false, b,
        /*c_mod=*/(short)0, c, /*reuse_a=*/false, /*reuse_b=*/false);
    return c;
}

// ---- K1: xl = x @ W_gcn^T (2->2) via WMMA, and deg init ----
__global__ void k1_xl_deg(const float* __restrict__ x, const float* __restrict__ Wg,
                          float* __restrict__ xl, float* __restrict__ deg) {
    const int wave = threadIdx.x >> 5;
    const int lane = threadIdx.x & 31;
    const int node_base = (blockIdx.x * 8 + wave) * 16;   // 1250 blocks -> exactly TN_
    const bool lo = lane < 16;
    const int  n  = lane & 15;
    const int  node = node_base + n;

    float a0 = lo ? x[node * 2 + 0] : 0.0f;
    float a1 = lo ? x[node * 2 + 1] : 0.0f;
    // B = W_gcn^T padded to 4x16: B[k][n] = W_gcn[n][k] for n<2,k<2
    float b0 = (lo && n < 2) ? Wg[n * 2 + 0] : 0.0f;
    float b1 = (lo && n < 2) ? Wg[n * 2 + 1] : 0.0f;

    v8f c = wmma_16x16x4(a0, a1, b0, b1);

    if (n < 2) {                       // lanes 0,1 (rows 0-7) and 16,17 (rows 8-15)
        const int mbase = lo ? 0 : 8;
#pragma unroll
        for (int r = 0; r < 8; ++r)
            xl[(node_base + mbase + r) * 2 + n] = c[r];
    }
    const int tid = blockIdx.x * blockDim.x + threadIdx.x;   // 320000 threads
    if (tid < TN_) deg[tid] = 1.0f;                          // self-loop
}

// ---- K2: degree accumulation over edges ----
__global__ void k2_degree(const int* __restrict__ ei, float* __restrict__ deg) {
    const int j  = blockIdx.x * blockDim.x + threadIdx.x;    // TE_/4 threads
    const int e0 = j * 4;                                    // never crosses a graph (E_%4==0)
    const int t  = e0 / E_;
    const int e  = e0 - t * E_;
    const int4 d4 = *(const int4*)(ei + t * 2 * E_ + E_ + e);
    float* dg = deg + t * N_;
    atomicAdd(&dg[d4.x], 1.0f);
    atomicAdd(&dg[d4.y], 1.0f);
    atomicAdd(&dg[d4.z], 1.0f);
    atomicAdd(&dg[d4.w], 1.0f);
}

// ---- K3: dinv = rsqrt(deg), in place ----
__global__ void k3_rsqrt(float* __restrict__ deg) {
    const int i = blockIdx.x * blockDim.x + threadIdx.x;
    if (i < TN_) deg[i] = rsqrtf(deg[i]);   // deg >= 1 always
}

// ---- K4: gout = b_gcn + dinv^2 * xl  (self-loop contribution) ----
__global__ void k4_selfinit(const float* __restrict__ dinv, const float* __restrict__ xl,
                            const float* __restrict__ bg, float* __restrict__ gout) {
    const int i = blockIdx.x * blockDim.x + threadIdx.x;
    if (i < TN_) {
        const float di = dinv[i];
        const float s  = di * di;
        gout[2 * i + 0] = __builtin_fmaf(s, xl[2 * i + 0], bg[0]);
        gout[2 * i + 1] = __builtin_fmaf(s, xl[2 * i + 1], bg[1]);
    }
}

__device__ inline void edge_accum(int s, int d, int base,
                                  const float* __restrict__ dinv,
                                  const float* __restrict__ xl,
                                  float* __restrict__ gout) {
    const float w = dinv[base + s] * dinv[base + d];
    const float vx = xl[2 * (base + s) + 0];
    const float vy = xl[2 * (base + s) + 1];
    atomicAdd(&gout[2 * (base + d) + 0], w * vx);
    atomicAdd(&gout[2 * (base + d) + 1], w * vy);
}

// ---- K5: weighted edge scatter ----
__global__ void k5_scatter(const int* __restrict__ ei, const float* __restrict__ dinv,
                           const float* __restrict__ xl, float* __restrict__ gout) {
    const int j  = blockIdx.x * blockDim.x + threadIdx.x;
    const int e0 = j * 4;
    const int t  = e0 / E_;
    const int e  = e0 - t * E_;
    const int4 s4 = *(const int4*)(ei + t * 2 * E_ + e);
    const int4 d4 = *(const int4*)(ei + t * 2 * E_ + E_ + e);
    const int base = t * N_;
    edge_accum(s4.x, d4.x, base, dinv, xl, gout);
    edge_accum(s4.y, d4.y, base, dinv, xl, gout);
    edge_accum(s4.z, d4.z, base, dinv, xl, gout);
    edge_accum(s4.w, d4.w, base, dinv, xl, gout);
}

// ---- K6: u = gout @ W_ih^T (2->3) + (b_ih + b_hh), via WMMA ----
__global__ void k6_inproj(const float* __restrict__ gout, const float* __restrict__ Wih,
                          const float* __restrict__ bih, const float* __restrict__ bhh,
                          float* __restrict__ u) {
    const int wave = threadIdx.x >> 5;
    const int lane = threadIdx.x & 31;
    const int node_base = (blockIdx.x * 8 + wave) * 16;   // 1250 blocks
    const bool lo = lane < 16;
    const int  n  = lane & 15;
    const int  node = node_base + n;

    float a0 = lo ? gout[node * 2 + 0] : 0.0f;
    float a1 = lo ? gout[node * 2 + 1] : 0.0f;
    // B = W_ih^T padded to 4x16: B[k][n] = W_ih[n][k] for n<3,k<2
    float b0 = (lo && n < 3) ? Wih[n * 2 + 0] : 0.0f;
    float b1 = (lo && n < 3) ? Wih[n * 2 + 1] : 0.0f;

    v8f c = wmma_16x16x4(a0, a1, b0, b1);

    if (n < 3) {                        // lanes 0-2 (rows 0-7), 16-18 (rows 8-15)
        const float bb = bih[n] + bhh[n];
        const int mbase = lo ? 0 : 8;
#pragma unroll
        for (int r = 0; r < 8; ++r)
            u[(node_base + mbase + r) * 3 + n] = c[r] + bb;
    }
}

__device__ inline float tanh_fast(float z) {
    z = fminf(10.0f, fmaxf(-10.0f, z));        // avoid inf/inf
    const float e = __expf(2.0f * z);          // v_exp_f32 path
    return 1.0f - __fdividef(2.0f, e + 1.0f);
}

// ---- K7: sequential RNN scan + head ----
__global__ void k7_scan(const float* __restrict__ u, const float* __restrict__ Whh,
                        const float* __restrict__ Wlin, const float* __restrict__ blin,
                        float* __restrict__ out) {
    if (threadIdx.x != 0 || blockIdx.x != 0) return;
    const float w00 = Whh[0], w01 = Whh[1], w02 = Whh[2];
    const float w10 = Whh[3], w11 = Whh[4], w12 = Whh[5];
    const float w20 = Whh[6], w21 = Whh[7], w22 = Whh[8];
    float h0 = 0.0f, h1 = 0.0f, h2 = 0.0f;
#pragma unroll 4
    for (int t = 0; t < TN_; ++t) {
        const float* ut = u + 3 * t;
        __builtin_prefetch(u + 3 * t + 768, 0, 3);   // global_prefetch_b8, ~256 steps ahead
        const float a0 = ut[0], a1 = ut[1], a2 = ut[2];
        const float z0 = __builtin_fmaf(w02, h2, __builtin_fmaf(w01, h1, __builtin_fmaf(w00, h0, a0)));
        const float z1 = __builtin_fmaf(w12, h2, __builtin_fmaf(w11, h1, __builtin_fmaf(w10, h0, a1)));
        const float z2 = __builtin_fmaf(w22, h2, __builtin_fmaf(w21, h1, __builtin_fmaf(w20, h0, a2)));
        h0 = tanh_fast(z0);
        h1 = tanh_fast(z1);
        h2 = tanh_fast(z2);
    }
    const float r0 = fmaxf(h0, 0.0f), r1 = fmaxf(h1, 0.0f), r2 = fmaxf(h2, 0.0f);
    float z = __builtin_fmaf(Wlin[2], r2,
              __builtin_fmaf(Wlin[1], r1,
              __builtin_fmaf(Wlin[0], r0, blin[0])));
    out[0] = __fdividef(1.0f, 1.0f + __expf(-z));
}

extern "C" void kernel_launch(void* const* d_in, const int* in_sizes, int n_in,
                              void* d_out, int out_size, void* d_ws, size_t ws_size,
                              hipStream_t stream) {
    (void)in_sizes; (void)n_in; (void)out_size; (void)ws_size;
    const float* x    = (const float*)d_in[0];
    const int*   ei   = (const int*)  d_in[1];   // int32 edge_index (T,2,E)
    const float* Wg   = (const float*)d_in[2];
    const float* bg   = (const float*)d_in[3];
    const float* Wih  = (const float*)d_in[4];
    const float* Whh  = (const float*)d_in[5];
    const float* bih  = (const float*)d_in[6];
    const float* bhh  = (const float*)d_in[7];
    const float* Wlin = (const float*)d_in[8];
    const float* blin = (const float*)d_in[9];
    float* out = (float*)d_out;

    float* ws   = (float*)d_ws;
    float* deg  = ws;            // TN  floats  (becomes dinv in place)
    float* xl   = ws + TN_;      // 2TN floats
    float* gout = ws + 3 * TN_;  // 2TN floats
    float* u    = ws + 5 * TN_;  // 3TN floats  -> 5.12 MB total

    const int edgeBlocks = (int)(TE_ / 4 / 256);   // 10000
    const int nodeBlocks = (TN_ + 255) / 256;      // 625
    const int wmmaBlocks = TN_ / 128;              // 1250 (full waves only)

    k1_xl_deg <<<wmmaBlocks, 256, 0, stream>>>(x, Wg, xl, deg);
    k2_degree <<<edgeBlocks, 256, 0, stream>>>(ei, deg);
    k3_rsqrt  <<<nodeBlocks, 256, 0, stream>>>(deg);
    k4_selfinit<<<nodeBlocks, 256, 0, stream>>>(deg, xl, bg, gout);
    k5_scatter<<<edgeBlocks, 256, 0, stream>>>(ei, deg, xl, gout);
    k6_inproj <<<wmmaBlocks, 256, 0, stream>>>(gout, Wih, bih, bhh, u);
    k7_scan   <<<1, 32, 0, stream>>>(u, Whh, Wlin, blin, out);
}